// VoxelGrid_15126874816752
// MI455X (gfx1250) — compile-verified
//
#include <hip/hip_runtime.h>

#define NUM_BINS 5

typedef __attribute__((ext_vector_type(4))) float v4f;

// ---------------------------------------------------------------------------
// Kernel 1: zero the output grid with pure 128-bit streaming stores.
// (Harness poisons d_out once and atomics accumulate, so we re-zero every
// launch. Default RT store hint keeps the freshly-zeroed lines allocated in
// L2, where the scatter atomics will then hit.)
// ---------------------------------------------------------------------------
__global__ __launch_bounds__(256) void voxel_zero_kernel(float4* __restrict__ out4,
                                                         int n4) {
    int i = blockIdx.x * blockDim.x + threadIdx.x;
    if (i < n4) {
        out4[i] = make_float4(0.0f, 0.0f, 0.0f, 0.0f);
    }
}

// Tail zeroing (only launched if out_size % 4 != 0 — never for this problem).
__global__ __launch_bounds__(64) void voxel_zero_tail_kernel(float* __restrict__ out,
                                                             int start, int n) {
    int i = start + blockIdx.x * blockDim.x + threadIdx.x;
    if (i < n) out[i] = 0.0f;
}

// ---------------------------------------------------------------------------
// Kernel 2: scatter events into the voxel grid.
//   - events are AoS float4 (t, x, y, p): one NON-TEMPORAL global_load_b128
//     per thread (a wave reads 512 contiguous bytes). NT keeps the 160 MB
//     stream from evicting the 37 MB L2-resident grid.
//   - two fire-and-forget global_atomic_add_f32 per event (no return value
//     -> no-return atomic encoding, STOREcnt only, no VGPR writeback).
//   - one global_prefetch_b8 per wave, ~8 KB ahead of the stream.
// ---------------------------------------------------------------------------
__global__ __launch_bounds__(256) void voxel_scatter_kernel(
    const v4f* __restrict__ ev, int n,
    const int* __restrict__ p_ct, const int* __restrict__ p_dt,
    const int* __restrict__ p_w,  const int* __restrict__ p_h,
    float* __restrict__ out) {

    int i = blockIdx.x * blockDim.x + threadIdx.x;
    if (i >= n) return;

    // One speculative prefetch per wave, ~8KB (512 events) ahead.
    if ((threadIdx.x & 31u) == 0u) {
        __builtin_prefetch((const char*)(ev + i) + 8192, 0, 0);
    }

    // Uniform scalar parameters (uniform address -> scalar loads).
    const int ct  = p_ct[0];
    const int dti = p_dt[0];
    const int W   = p_w[0];
    const int H   = p_h[0];

    const float bt     = (float)(ct - dti);
    const float inv_dt = 1.0f / (float)(dti + 1);   // DT_OFFSET = 1

    // Non-temporal 128-bit load of (t, x, y, p).
    v4f e = __builtin_nontemporal_load(ev + i);
    const float t = e.x;
    const float x = e.y;
    const float y = e.z;
    const float p = e.w;

    const float r     = (t - bt) * inv_dt;            // tn in [0,1)
    const float bin_f = (float)(NUM_BINS - 1) * r;    // in [0, 4)
    const float back  = floorf(bin_f);
    const float fw    = bin_f - back;                 // forward weight

    // bin_f >= 0 by construction; guard only the upper edge (fp safety).
    const int back_i = min((int)back, NUM_BINS - 2);

    const int pos   = (p > 0.0f) ? 1 : 0;             // KEEP_POLARITY
    const int xi    = (int)x;
    const int yi    = (int)y;
    const int plane = H * W;

    const int idx = ((back_i * 2 + pos) * H + yi) * W + xi;

    // backward-bin contribution: (1-fw)*tn ; forward bin is exactly +2 planes.
    atomicAdd(out + idx,              (1.0f - fw) * r);
    atomicAdd(out + idx + 2 * plane,  fw * r);
}

// ---------------------------------------------------------------------------
// Launch
// ---------------------------------------------------------------------------
extern "C" void kernel_launch(void* const* d_in, const int* in_sizes, int n_in,
                              void* d_out, int out_size, void* d_ws, size_t ws_size,
                              hipStream_t stream) {
    const float* events = (const float*)d_in[0];   // (N,4) row-major: t,x,y,p
    const int*   p_ct   = (const int*)d_in[1];     // curr_time
    const int*   p_dt   = (const int*)d_in[2];     // delta_t
    const int*   p_w    = (const int*)d_in[3];     // width
    const int*   p_h    = (const int*)d_in[4];     // height
    float*       out    = (float*)d_out;

    const int n  = in_sizes[0] / 4;                // number of events
    const int n4 = out_size / 4;

    // Zero the grid (b128 stores), then scatter. Same stream => ordered.
    if (n4 > 0) {
        voxel_zero_kernel<<<(n4 + 255) / 256, 256, 0, stream>>>((float4*)d_out, n4);
    }
    if (out_size % 4 != 0) {  // dead for this problem; host-side decision
        voxel_zero_tail_kernel<<<1, 64, 0, stream>>>(out, n4 * 4, out_size);
    }
    voxel_scatter_kernel<<<(n + 255) / 256, 256, 0, stream>>>(
        (const v4f*)events, n, p_ct, p_dt, p_w, p_h, out);
}